// LogSqrt2Quantizer_46024869544618
// MI455X (gfx1250) — compile-verified
//
#include <hip/hip_runtime.h>
#include <hip/hip_bf16.h>
#include <stdint.h>

// ---- compile-safe probes for gfx1250 async-memory builtins (info via #warning in stderr) ----
#if __has_builtin(__builtin_amdgcn_global_load_async_to_lds_b128)
#warning "PROBE: HAS __builtin_amdgcn_global_load_async_to_lds_b128"
#endif
#if __has_builtin(__builtin_amdgcn_global_store_async_from_lds_b128)
#warning "PROBE: HAS __builtin_amdgcn_global_store_async_from_lds_b128"
#endif
#if __has_builtin(__builtin_amdgcn_s_wait_asynccnt)
#warning "PROBE: HAS __builtin_amdgcn_s_wait_asynccnt"
#endif

typedef float v4f __attribute__((ext_vector_type(4)));

#define INT_MAXV  0x7fffffff
#define INT_MINV  (-0x7fffffff - 1)

__device__ __forceinline__ int rlog(float x, float sx, float bias) {
    // x_int = round(x / s_x) + bias ; r = round(log2(x_int))   (RNE, IEEE div: matches jnp)
    float xi = rintf(x / sx) + bias;
    return (int)rintf(log2f(xi));
}

// ---------------------------------------------------------------------------
// init: reset min/max accumulators each launch (harness does not re-poison ws
// between graph replays) and emit the trailing scalar output s_x.
// ---------------------------------------------------------------------------
__global__ void lsq_init(int* __restrict__ mm, const float* __restrict__ sxp,
                         float* __restrict__ tail_scalar) {
    mm[0] = INT_MAXV;  // min of r
    mm[1] = INT_MINV;  // max of r
    *tail_scalar = *sxp;
}

// ---------------------------------------------------------------------------
// shared: build the r -> final-output LUT in LDS.  Everything downstream of
// r is a pure per-block-uniform function of r, reconstructed with the exact
// fp32 op sequence the reference applies elementwise.
// ---------------------------------------------------------------------------
__device__ __forceinline__ int build_lut(float* lut, int rmin, int rmax,
                                         float sx, float t2,
                                         const float* __restrict__ unq, int n_unq,
                                         const float* __restrict__ mapv, int n_map) {
    int lutN = rmax - rmin + 1;
    if (lutN < 1)  lutN = 1;
    if (lutN > 64) lutN = 64;
    if ((int)threadIdx.x < lutN) {
        const float mn = -(float)rmax * t2;           // == xlq.min()
        const float mx = -(float)rmin * t2;           // == xlq.max()
        const float scale = (mx - mn) / 15.0f;        // N_LEVELS-1
        const float zp = rintf(-mn / scale);
        const float r = (float)(rmin + (int)threadIdx.x);
        const float xlq = -r * t2;
        float q = rintf(xlq / scale) + zp;
        q = fminf(fmaxf(q, 0.0f), 15.0f);
        const float dq = rintf((q - zp) * scale);
        float o = dq;
        const int nrm = (n_unq < n_map) ? n_unq : n_map;
        for (int i = 0; i < nrm; ++i) {               // sequential where-ordering preserved
            if (o == unq[i]) o = mapv[n_map - 1 - i];
        }
        lut[threadIdx.x] = o * sx;
    }
    __syncthreads();
    return lutN;
}

// ===========================================================================
// PATH A: r-byte cache through L2.
// ===========================================================================

// pass 1: read x (NT b128), compute r, reduce min/max, store r packed 1B/elem
// with DEFAULT (RT) policy so the 100MB intermediate parks in the 192MB L2.
__global__ void __launch_bounds__(256)
lsq_p1_store(const v4f* __restrict__ x4, size_t nv,
             const float* __restrict__ xt, int ntail,
             const float* __restrict__ sxp, const float* __restrict__ biasp,
             int* __restrict__ mm, uint32_t* __restrict__ rpack) {
    __shared__ int smin, smax;
    if (threadIdx.x == 0) { smin = INT_MAXV; smax = INT_MINV; }
    __syncthreads();
    const float sx = *sxp;
    const float bias = *biasp;
    int lmin = INT_MAXV, lmax = INT_MINV;
    const size_t stride = (size_t)gridDim.x * blockDim.x;
    for (size_t i = (size_t)blockIdx.x * blockDim.x + threadIdx.x; i < nv; i += stride) {
        v4f v = __builtin_nontemporal_load(&x4[i]);   // stream once: 400MB >> L2
        uint32_t pk = 0;
#pragma unroll
        for (int k = 0; k < 4; ++k) {
            int r = rlog(v[k], sx, bias);
            lmin = min(lmin, r);
            lmax = max(lmax, r);
            int rb = r < 0 ? 0 : (r > 255 ? 255 : r);
            pk |= (uint32_t)rb << (8 * k);
        }
        rpack[i] = pk;                                // RT store -> L2-resident
    }
    if (blockIdx.x == 0 && threadIdx.x == 0) {
        for (int k = 0; k < ntail; ++k) {             // tail folded into min/max only
            int r = rlog(xt[k], sx, bias);
            lmin = min(lmin, r);
            lmax = max(lmax, r);
        }
    }
    atomicMin(&smin, lmin);
    atomicMax(&smax, lmax);
    __syncthreads();
    if (threadIdx.x == 0) { atomicMin(&mm[0], smin); atomicMax(&mm[1], smax); }
}

// pass 2: consume r with th:TH_LOAD_LU (last-use: NT + discard-dirty-on-hit),
// so the L2-resident intermediate is never written back to HBM.  Then a pure
// LDS-LUT gather and NT b128 store of the final output.
__global__ void __launch_bounds__(256)
lsq_p2_lut(const uint32_t* __restrict__ rpack, size_t nv,
           const float* __restrict__ xt, float* __restrict__ ot, int ntail,
           const float* __restrict__ sxp, const float* __restrict__ biasp,
           const float* __restrict__ timesp,
           const float* __restrict__ unq, int n_unq,
           const float* __restrict__ mapv, int n_map,
           const int* __restrict__ mm,
           v4f* __restrict__ out4) {
    __shared__ float lut[64];
    const int rmin = mm[0];
    const int rmax = mm[1];
    const float sx = *sxp;
    const float bias = *biasp;
    const float t2 = exp2f(*timesp);
    const int lutN = build_lut(lut, rmin, rmax, sx, t2, unq, n_unq, mapv, n_map);

    const size_t i = (size_t)blockIdx.x * blockDim.x + threadIdx.x;
    if (i < nv) {
        const uint32_t* ap = rpack + i;
        uint32_t pk;
        asm volatile("global_load_b32 %0, %1, off th:TH_LOAD_LU\n\t"
                     "s_wait_loadcnt 0x0"
                     : "=v"(pk) : "v"(ap) : "memory");
        v4f o;
#pragma unroll
        for (int k = 0; k < 4; ++k) {
            int idx = (int)((pk >> (8 * k)) & 0xffu) - rmin;
            idx = idx < 0 ? 0 : (idx >= lutN ? lutN - 1 : idx);
            o[k] = lut[idx];
        }
        __builtin_nontemporal_store(o, &out4[i]);
    }
    if (blockIdx.x == 0 && threadIdx.x == 0) {        // tail: recompute from x directly
        for (int k = 0; k < ntail; ++k) {
            int idx = rlog(xt[k], sx, bias) - rmin;
            idx = idx < 0 ? 0 : (idx >= lutN ? lutN - 1 : idx);
            ot[k] = lut[idx];
        }
    }
}

// ===========================================================================
// PATH B fallback (workspace too small): recompute r in pass 2.
// ===========================================================================
__global__ void __launch_bounds__(256)
lsq_minmax(const v4f* __restrict__ x4, size_t nv,
           const float* __restrict__ xt, int ntail,
           const float* __restrict__ sxp, const float* __restrict__ biasp,
           int* __restrict__ mm) {
    __shared__ int smin, smax;
    if (threadIdx.x == 0) { smin = INT_MAXV; smax = INT_MINV; }
    __syncthreads();
    const float sx = *sxp;
    const float bias = *biasp;
    int lmin = INT_MAXV, lmax = INT_MINV;
    const size_t stride = (size_t)gridDim.x * blockDim.x;
    for (size_t i = (size_t)blockIdx.x * blockDim.x + threadIdx.x; i < nv; i += stride) {
        v4f v = __builtin_nontemporal_load(&x4[i]);
#pragma unroll
        for (int k = 0; k < 4; ++k) {
            int r = rlog(v[k], sx, bias);
            lmin = min(lmin, r);
            lmax = max(lmax, r);
        }
    }
    if (blockIdx.x == 0 && threadIdx.x == 0) {
        for (int k = 0; k < ntail; ++k) {
            int r = rlog(xt[k], sx, bias);
            lmin = min(lmin, r);
            lmax = max(lmax, r);
        }
    }
    atomicMin(&smin, lmin);
    atomicMax(&smax, lmax);
    __syncthreads();
    if (threadIdx.x == 0) { atomicMin(&mm[0], smin); atomicMax(&mm[1], smax); }
}

__global__ void __launch_bounds__(256)
lsq_apply(const v4f* __restrict__ x4, size_t nv,
          const float* __restrict__ xt, float* __restrict__ ot, int ntail,
          const float* __restrict__ sxp, const float* __restrict__ biasp,
          const float* __restrict__ timesp,
          const float* __restrict__ unq, int n_unq,
          const float* __restrict__ mapv, int n_map,
          const int* __restrict__ mm,
          v4f* __restrict__ out4) {
    __shared__ float lut[64];
    const int rmin = mm[0];
    const int rmax = mm[1];
    const float sx = *sxp;
    const float bias = *biasp;
    const float t2 = exp2f(*timesp);
    const int lutN = build_lut(lut, rmin, rmax, sx, t2, unq, n_unq, mapv, n_map);

    const size_t i = (size_t)blockIdx.x * blockDim.x + threadIdx.x;
    if (i < nv) {
        v4f v = __builtin_nontemporal_load(&x4[i]);
        v4f o;
#pragma unroll
        for (int k = 0; k < 4; ++k) {
            int idx = rlog(v[k], sx, bias) - rmin;
            idx = idx < 0 ? 0 : (idx >= lutN ? lutN - 1 : idx);
            o[k] = lut[idx];
        }
        __builtin_nontemporal_store(o, &out4[i]);
    }
    if (blockIdx.x == 0 && threadIdx.x == 0) {
        for (int k = 0; k < ntail; ++k) {
            int idx = rlog(xt[k], sx, bias) - rmin;
            idx = idx < 0 ? 0 : (idx >= lutN ? lutN - 1 : idx);
            ot[k] = lut[idx];
        }
    }
}

// ---------------------------------------------------------------------------
// launch.  Inputs (setup_inputs order):
//   [0] x_hat (N f32) [1] s_x (1) [2] unq [3] map_vals [4] int_bias (1) [5] times (1)
// d_out: N floats (out*s_x) + 1 float (s_x).
// d_ws layout: [0,8) int min/max ; [256, 256+N) packed r bytes (path A).
// ---------------------------------------------------------------------------
extern "C" void kernel_launch(void* const* d_in, const int* in_sizes, int n_in,
                              void* d_out, int out_size, void* d_ws, size_t ws_size,
                              hipStream_t stream) {
    const float* x     = (const float*)d_in[0];
    const float* sx    = (const float*)d_in[1];
    const float* unq   = (const float*)d_in[2];
    const float* mapv  = (const float*)d_in[3];
    const float* biasp = (const float*)d_in[4];
    const float* times = (const float*)d_in[5];
    const int n_unq = in_sizes[2];
    const int n_map = in_sizes[3];
    const size_t n = (size_t)in_sizes[0];
    float* out = (float*)d_out;

    uint8_t* wsb = (uint8_t*)d_ws;
    int* mm = (int*)wsb;
    uint32_t* rpack = (uint32_t*)(wsb + 256);

    const size_t nv = n / 4;
    const int ntail = (int)(n - nv * 4);
    const bool cache_r = (ws_size >= nv * 4 + 256);

    lsq_init<<<1, 1, 0, stream>>>(mm, sx, out + n);

    size_t b1 = (nv + 255) / 256;
    if (b1 > 6144) b1 = 6144;            // grid-stride; few atomics per block
    if (b1 < 1) b1 = 1;
    size_t b2 = (nv + 255) / 256;
    if (b2 < 1) b2 = 1;

    if (cache_r) {
        lsq_p1_store<<<(int)b1, 256, 0, stream>>>((const v4f*)x, nv, x + nv * 4, ntail,
                                                  sx, biasp, mm, rpack);
        lsq_p2_lut<<<(int)b2, 256, 0, stream>>>(rpack, nv, x + nv * 4, out + nv * 4, ntail,
                                                sx, biasp, times, unq, n_unq, mapv, n_map,
                                                mm, (v4f*)out);
    } else {
        lsq_minmax<<<(int)b1, 256, 0, stream>>>((const v4f*)x, nv, x + nv * 4, ntail,
                                                sx, biasp, mm);
        lsq_apply<<<(int)b2, 256, 0, stream>>>((const v4f*)x, nv, x + nv * 4, out + nv * 4, ntail,
                                               sx, biasp, times, unq, n_unq, mapv, n_map,
                                               mm, (v4f*)out);
    }
}